// GCN_scheduling_4501125726340
// MI455X (gfx1250) — compile-verified
//
#include <hip/hip_runtime.h>
#include <math.h>

typedef __attribute__((ext_vector_type(2))) float v2f;
typedef __attribute__((ext_vector_type(8))) float v8f;

#define HEADS 5
#define C1HID 32
#define F1 160   // HEADS * C1HID

// ---------- float atomic max via sign-split int atomics ----------
__device__ __forceinline__ void atomicMaxF(float* addr, float v) {
  if (v >= 0.0f) atomicMax((int*)addr, __float_as_int(v));
  else           atomicMin((unsigned int*)addr, __float_as_uint(v));
}

// ---------- generic fill ----------
__global__ void k_fill(float* p, long n, float v) {
  long i = (long)blockIdx.x * blockDim.x + threadIdx.x;
  long stride = (long)gridDim.x * blockDim.x;
  for (; i < n; i += stride) p[i] = v;
}

// ---------- sum(edge_attr) reduction ----------
__global__ void k_easum(const float* __restrict__ ea, int E, float* sum) {
  __shared__ float sm[256];
  float acc = 0.f;
  for (long i = (long)blockIdx.x * blockDim.x + threadIdx.x; i < E;
       i += (long)gridDim.x * blockDim.x)
    acc += ea[i];
  sm[threadIdx.x] = acc;
  __syncthreads();
  for (int s = 128; s > 0; s >>= 1) {
    if ((int)threadIdx.x < s) sm[threadIdx.x] += sm[threadIdx.x + s];
    __syncthreads();
  }
  if (threadIdx.x == 0) atomicAdd(sum, sm[0]);
}

// ---------- per-head edge constants: consts[0]=ea_mean, [1..5]=ce1, [6..10]=ce2 ----------
__global__ void k_consts(const float* __restrict__ easum, int E,
                         const float* __restrict__ We1, const float* __restrict__ ae1,
                         const float* __restrict__ We2, const float* __restrict__ ae2,
                         float* consts) {
  if (threadIdx.x != 0 || blockIdx.x != 0) return;
  consts[0] = easum[0] / (float)E;
  for (int h = 0; h < HEADS; ++h) {
    float s = 0.f;
    for (int c = 0; c < C1HID; ++c) s += We1[h * C1HID + c] * ae1[h * C1HID + c];
    consts[1 + h] = s;
    consts[6 + h] = We2[h] * ae2[h];
  }
}

// ---------- layer-1 GEMM: h1[N,160] = x[N,5] @ W1[5,160] via V_WMMA_F32_16X16X4_F32 ----------
// One wave per 16x16 output tile. K=5 handled as WMMA(K=4) + VALU tail for K=4 column.
__global__ __launch_bounds__(32) void k_gemm1_wmma(const float* __restrict__ x,
                                                   const float* __restrict__ W1,
                                                   float* __restrict__ h1) {
  int tile = blockIdx.x;
  int rt = tile / 10, ct = tile % 10;
  int lane = threadIdx.x;
  int half = lane >> 4, l = lane & 15;
  int r0 = rt * 16, c0 = ct * 16;
  // A 16x4 layout: lanes 0-15 hold K=0,1 ; lanes 16-31 hold K=2,3 (per row M=l)
  v2f a, b;
  const float* xr = x + (long)(r0 + l) * 5 + half * 2;
  a.x = xr[0];
  a.y = xr[1];
  // B 4x16 layout (mirror): V0 = rows K=0 (lanes 0-15) / K=2 (lanes 16-31); V1 = K=1 / K=3
  b.x = W1[(half * 2 + 0) * F1 + c0 + l];
  b.y = W1[(half * 2 + 1) * F1 + c0 + l];
  v8f c = {};
  c = __builtin_amdgcn_wmma_f32_16x16x4_f32(false, a, false, b, (short)0, c, false, false);
  // K=4 tail + store. D layout: VGPR i holds row M=i (lanes 0-15) / M=i+8 (lanes 16-31)
  float w4 = W1[4 * F1 + c0 + l];
#pragma unroll
  for (int i = 0; i < 8; ++i) {
    int row = r0 + i + half * 8;
    float v = c[i] + x[(long)row * 5 + 4] * w4;
    h1[(long)row * F1 + c0 + l] = v;
  }
}

// ---------- layer-2 GEMM: h2[N,5] = x2[N,160] @ W2[160,5], K-loop of WMMA f32 16x16x4 ----------
__global__ __launch_bounds__(32) void k_gemm2_wmma(const float* __restrict__ x2,
                                                   const float* __restrict__ W2,
                                                   float* __restrict__ h2) {
  int rt = blockIdx.x;
  int lane = threadIdx.x;
  int half = lane >> 4, l = lane & 15;
  int r0 = rt * 16;
  bool valid = (l < HEADS);  // only 5 of 16 output columns are real
  v8f c = {};
  for (int k = 0; k < F1; k += 4) {
    v2f a, b;
    const float* xr = x2 + (long)(r0 + l) * F1 + k + half * 2;
    a.x = xr[0];
    a.y = xr[1];
    int k0 = k + half * 2;
    b.x = valid ? W2[(k0 + 0) * HEADS + l] : 0.f;
    b.y = valid ? W2[(k0 + 1) * HEADS + l] : 0.f;
    c = __builtin_amdgcn_wmma_f32_16x16x4_f32(false, a, false, b, (short)0, c, false, false);
  }
  if (valid) {
#pragma unroll
    for (int i = 0; i < 8; ++i)
      h2[(long)(r0 + i + half * 8) * HEADS + l] = c[i];
  }
}

// ---------- layer-1 attention projections: [N,5] dots over C=32 ----------
__global__ void k_alpha1(const float* __restrict__ h1, const float* __restrict__ a_src,
                         const float* __restrict__ a_dst, int N,
                         float* __restrict__ as, float* __restrict__ ad) {
  int i = blockIdx.x * blockDim.x + threadIdx.x;
  if (i >= N * HEADS) return;
  int n = i / HEADS, h = i % HEADS;
  const float* hp = h1 + (long)n * F1 + h * C1HID;
  float ss = 0.f, sd = 0.f;
#pragma unroll
  for (int c = 0; c < C1HID; ++c) {
    float v = hp[c];
    ss += v * a_src[h * C1HID + c];
    sd += v * a_dst[h * C1HID + c];
  }
  as[i] = ss;
  ad[i] = sd;
}

// ---------- layer-2 attention projections (C=1) ----------
__global__ void k_alpha2(const float* __restrict__ h2, const float* __restrict__ a_src,
                         const float* __restrict__ a_dst, int N,
                         float* __restrict__ as, float* __restrict__ ad) {
  int i = blockIdx.x * blockDim.x + threadIdx.x;
  if (i >= N * HEADS) return;
  int h = i % HEADS;
  float v = h2[i];
  as[i] = v * a_src[h];
  ad[i] = v * a_dst[h];
}

// ---------- edge logits + segment max (shared by both layers) ----------
__global__ void k_edge_logits(const float* __restrict__ as, const float* __restrict__ ad,
                              const float* __restrict__ ea, const int* __restrict__ src,
                              const int* __restrict__ dst, const float* __restrict__ consts,
                              int E, int N, int ce_off,
                              float* __restrict__ logits, float* __restrict__ m) {
  int e = blockIdx.x * blockDim.x + threadIdx.x;
  if (e >= E + N) return;
  int s, d;
  float eav;
  if (e < E) { s = src[e]; d = dst[e]; eav = ea[e]; }
  else       { s = d = e - E; eav = consts[0]; }  // self-loop, mean edge_attr
#pragma unroll
  for (int h = 0; h < HEADS; ++h) {
    float z = as[s * HEADS + h] + ad[d * HEADS + h] + eav * consts[ce_off + h];
    z = z > 0.f ? z : 0.2f * z;  // leaky_relu(0.2)
    logits[(long)e * HEADS + h] = z;
    atomicMaxF(&m[d * HEADS + h], z);
  }
}

// ---------- exp(logit - max) + segment sum (in-place logits -> ex) ----------
__global__ void k_edge_expsum(const int* __restrict__ dst, int E, int N,
                              const float* __restrict__ m, float* __restrict__ logits,
                              float* __restrict__ denom) {
  int e = blockIdx.x * blockDim.x + threadIdx.x;
  if (e >= E + N) return;
  int d = (e < E) ? dst[e] : (e - E);
#pragma unroll
  for (int h = 0; h < HEADS; ++h) {
    float ex = __expf(logits[(long)e * HEADS + h] - m[d * HEADS + h]);
    logits[(long)e * HEADS + h] = ex;
    atomicAdd(&denom[d * HEADS + h], ex);
  }
}

// ---------- layer-1 message scatter: one wave per edge, lane = channel within head ----------
__global__ void k_scatter1(const float* __restrict__ ex, const float* __restrict__ denom,
                           const float* __restrict__ h1, const int* __restrict__ src,
                           const int* __restrict__ dst, int E, int N,
                           float* __restrict__ agg) {
  int wave = (int)(((long)blockIdx.x * blockDim.x + threadIdx.x) >> 5);
  int lane = threadIdx.x & 31;
  if (wave >= E + N) return;
  int s, d;
  if (wave < E) { s = src[wave]; d = dst[wave]; }
  else          { s = d = wave - E; }
#pragma unroll
  for (int h = 0; h < HEADS; ++h) {
    float a = ex[(long)wave * HEADS + h] / (denom[d * HEADS + h] + 1e-16f);
    float v = a * h1[(long)s * F1 + h * C1HID + lane];
    atomicAdd(&agg[(long)d * F1 + h * C1HID + lane], v);
  }
}

// ---------- layer-2 message scatter (C=1): thread per edge ----------
__global__ void k_scatter2(const float* __restrict__ ex, const float* __restrict__ denom,
                           const float* __restrict__ h2, const int* __restrict__ src,
                           const int* __restrict__ dst, int E, int N,
                           float* __restrict__ agg2) {
  int e = blockIdx.x * blockDim.x + threadIdx.x;
  if (e >= E + N) return;
  int s, d;
  if (e < E) { s = src[e]; d = dst[e]; }
  else       { s = d = e - E; }
#pragma unroll
  for (int h = 0; h < HEADS; ++h) {
    float a = ex[(long)e * HEADS + h] / (denom[d * HEADS + h] + 1e-16f);
    atomicAdd(&agg2[d * HEADS + h], a * h2[s * HEADS + h]);
  }
}

// ---------- x2 = relu(agg1 + b1), in place ----------
__global__ void k_relu_bias(float* __restrict__ x2, const float* __restrict__ b1, long n) {
  long i = (long)blockIdx.x * blockDim.x + threadIdx.x;
  if (i >= n) return;
  float v = x2[i] + b1[i % F1];
  x2[i] = v > 0.f ? v : 0.f;
}

// ---------- out = sigmoid((mean_h agg2 + b2) * Wlin) ----------
__global__ void k_final(const float* __restrict__ agg2, const float* __restrict__ b2,
                        const float* __restrict__ Wlin, int N, float* __restrict__ out) {
  int n = blockIdx.x * blockDim.x + threadIdx.x;
  if (n >= N) return;
  float s = 0.f;
#pragma unroll
  for (int h = 0; h < HEADS; ++h) s += agg2[n * HEADS + h];
  float z = (s * (1.0f / HEADS) + b2[0]) * Wlin[0];
  out[n] = 1.0f / (1.0f + __expf(-z));
}

extern "C" void kernel_launch(void* const* d_in, const int* in_sizes, int n_in,
                              void* d_out, int out_size, void* d_ws, size_t ws_size,
                              hipStream_t stream) {
  const float* x        = (const float*)d_in[0];
  const float* eattr    = (const float*)d_in[1];
  const int*   src      = (const int*)d_in[2];
  const int*   dst      = (const int*)d_in[3];
  const float* W1       = (const float*)d_in[4];
  const float* a_src1   = (const float*)d_in[5];
  const float* a_dst1   = (const float*)d_in[6];
  const float* We1      = (const float*)d_in[7];
  const float* ae1      = (const float*)d_in[8];
  const float* b1       = (const float*)d_in[9];
  const float* W2       = (const float*)d_in[10];
  const float* a_src2   = (const float*)d_in[11];
  const float* a_dst2   = (const float*)d_in[12];
  const float* We2      = (const float*)d_in[13];
  const float* ae2      = (const float*)d_in[14];
  const float* b2       = (const float*)d_in[15];
  const float* Wlin     = (const float*)d_in[16];
  float* out = (float*)d_out;

  const int N = in_sizes[0] / 5;   // 50000 (multiple of 16)
  const int E = in_sizes[2];       // 800000
  const int Etot = E + N;

  // ---- workspace layout (floats) ----
  float* ws = (float*)d_ws;
  size_t off = 0;
  float* h1    = ws + off; off += (size_t)N * F1;          // [N,160]
  float* logit = ws + off; off += (size_t)Etot * HEADS;    // logits/ex, reused by both layers
  float* as1   = ws + off; off += (size_t)N * HEADS;
  float* ad1   = ws + off; off += (size_t)N * HEADS;
  float* as2   = ws + off; off += (size_t)N * HEADS;
  float* ad2   = ws + off; off += (size_t)N * HEADS;
  float* h2    = ws + off; off += (size_t)N * HEADS;
  // zero-init block (contiguous)
  float* zblk  = ws + off;
  float* agg1  = ws + off; off += (size_t)N * F1;          // becomes x2 after relu+bias
  float* agg2  = ws + off; off += (size_t)N * HEADS;
  float* den1  = ws + off; off += (size_t)N * HEADS;
  float* den2  = ws + off; off += (size_t)N * HEADS;
  float* easum = ws + off; off += 1;
  float* consts= ws + off; off += 16;                      // [0]=ea_mean, [1..5]=ce1, [6..10]=ce2
  long  zcount = (long)((ws + off) - zblk);
  // -inf block (contiguous)
  float* nblk  = ws + off;
  float* m1    = ws + off; off += (size_t)N * HEADS;
  float* m2    = ws + off; off += (size_t)N * HEADS;
  long  ncount = (long)((ws + off) - nblk);

  const int TB = 256;
  const float NEG_INF = -__builtin_huge_valf();

  // 1) init
  k_fill<<<4096, TB, 0, stream>>>(zblk, zcount, 0.0f);
  k_fill<<<1024, TB, 0, stream>>>(nblk, ncount, NEG_INF);
  // 2) edge_attr mean + per-head edge constants
  k_easum<<<512, TB, 0, stream>>>(eattr, E, easum);
  k_consts<<<1, 32, 0, stream>>>(easum, E, We1, ae1, We2, ae2, consts);
  // 3) layer-1 dense projection (WMMA)
  k_gemm1_wmma<<<(N / 16) * 10, 32, 0, stream>>>(x, W1, h1);
  k_alpha1<<<(N * HEADS + TB - 1) / TB, TB, 0, stream>>>(h1, a_src1, a_dst1, N, as1, ad1);
  // 4) layer-1 segment softmax + aggregate
  k_edge_logits<<<(Etot + TB - 1) / TB, TB, 0, stream>>>(as1, ad1, eattr, src, dst, consts,
                                                         E, N, 1, logit, m1);
  k_edge_expsum<<<(Etot + TB - 1) / TB, TB, 0, stream>>>(dst, E, N, m1, logit, den1);
  k_scatter1<<<((long)Etot * 32 + TB - 1) / TB, TB, 0, stream>>>(logit, den1, h1, src, dst,
                                                                 E, N, agg1);
  // 5) relu + bias (agg1 -> x2 in place)
  k_relu_bias<<<((long)N * F1 + TB - 1) / TB, TB, 0, stream>>>(agg1, b1, (long)N * F1);
  // 6) layer-2 dense projection (WMMA K-loop)
  k_gemm2_wmma<<<N / 16, 32, 0, stream>>>(agg1, W2, h2);
  k_alpha2<<<(N * HEADS + TB - 1) / TB, TB, 0, stream>>>(h2, a_src2, a_dst2, N, as2, ad2);
  // 7) layer-2 segment softmax + aggregate
  k_edge_logits<<<(Etot + TB - 1) / TB, TB, 0, stream>>>(as2, ad2, eattr, src, dst, consts,
                                                         E, N, 6, logit, m2);
  k_edge_expsum<<<(Etot + TB - 1) / TB, TB, 0, stream>>>(dst, E, N, m2, logit, den2);
  k_scatter2<<<(Etot + TB - 1) / TB, TB, 0, stream>>>(logit, den2, h2, src, dst, E, N, agg2);
  // 8) head-mean + linear + sigmoid
  k_final<<<(N + TB - 1) / TB, TB, 0, stream>>>(agg2, b2, Wlin, N, out);
}